// CF_DCN_75617194213554
// MI455X (gfx1250) — compile-verified
//
#include <hip/hip_runtime.h>
#include <math.h>

// Problem constants
#define BATCH 8
#define CIN   64
#define CIN2  66
#define HH    128
#define WW    128
#define COUT  64
#define CHOM  27
#define KDIM  594              // CIN2 * 9
#define OUT_OFF  0
#define IDX_OFF  (BATCH*COUT*HH*WW)              // 8,388,608
#define MASK_OFF (IDX_OFF + BATCH*18*HH*WW)      // 10,747,904

typedef __attribute__((ext_vector_type(2))) float v2f;
typedef __attribute__((ext_vector_type(8))) float v8f;

#define WMMA_F32(ACC, AV, BV) \
  ACC = __builtin_amdgcn_wmma_f32_16x16x4_f32(false, AV, false, BV, (short)0, ACC, false, false)

// K-space permutation (shared by A-stager and B-builder):
//   new-k in [0,576)  : t = k>>6 (tap), c = k&63          (real channels)
//   new-k in [576,594): j = k-576, t = j>>1, c = 64+(j&1) (coord channels)
//   new-k in [594,640): zero padding
// original weight index = m*594 + c*9 + t.

// ===========================================================================
// Kernel 1: offset/mask conv as implicit GEMM on the matrix pipe.
//   M = 27 (pad 32 -> 2 tiles), N = 16-pixel strip/wave, K = 594 (pad 640).
// ===========================================================================
__global__ __launch_bounds__(256) void dcn_om_wmma_kernel(
    const float* __restrict__ input, const float* __restrict__ w_om,
    const float* __restrict__ b_om, float* __restrict__ out) {
  __shared__ float s_w[32 * 64];             // w_om chunk [m][kk], 8 KB

  const int tid  = threadIdx.x;
  const int wave = tid >> 5;
  const int lane = tid & 31;
  const int g    = lane >> 4;
  const int px   = lane & 15;

  const int strip = blockIdx.x * 8 + wave;   // 8192 strips
  const int wtile = strip & 7;
  const int ho    = (strip >> 3) & 127;
  const int b     = strip >> 10;
  const int wo    = (wtile << 4) + px;
  const float inv128 = 1.f / 128.f;

  const float* inBase = input + ((size_t)b * CIN << 14);

  v8f acc0 = {}; v8f acc1 = {};

  for (int kb = 0; kb < 10; ++kb) {
    __syncthreads();
    // Stage permuted w_om chunk: s_w row m, col kk holds w_om for output m, channel c(kk), tap t(kk)
    for (int i = tid; i < 2048; i += 256) {
      int m = i >> 6, kk = i & 63;
      float v = 0.f;
      if (m < CHOM) {
        if (kb < 9)              v = w_om[m * KDIM + kk * 9 + kb];
        else if (kk < 18)        v = w_om[m * KDIM + (CIN + (kk & 1)) * 9 + (kk >> 1)];
      }
      s_w[i] = v;
    }
    __syncthreads();

    if (kb < 9) {
      // tap t = kb fixed for whole block: hoist coords + bounds mask
      int h2 = ho + kb / 3 - 1;
      int w2 = wo + kb % 3 - 1;
      bool ok = (h2 >= 0) & (h2 < HH) & (w2 >= 0) & (w2 < WW);
      float okf = ok ? 1.f : 0.f;
      const float* xb = inBase + ((min(max(h2, 0), HH - 1) << 7) + min(max(w2, 0), WW - 1));
#pragma unroll 4
      for (int s = 0; s < 16; ++s) {
        int c0 = 4 * s + 2 * g;
        v2f bv = {xb[(size_t)c0 << 14] * okf,
                  xb[(size_t)(c0 + 1) << 14] * okf};
        int kk0 = 4 * s + 2 * g;
        v2f a0 = *(const v2f*)&s_w[((0 * 16 + px) << 6) + kk0];
        v2f a1 = *(const v2f*)&s_w[((1 * 16 + px) << 6) + kk0];
        WMMA_F32(acc0, a0, bv);
        WMMA_F32(acc1, a1, bv);
      }
    } else {
      // analytic coord channels: j = 4s+2g+{0,1}, t=j>>1, c=64+(j&1)
      for (int s = 0; s < 16; ++s) {
        int j0 = 4 * s + 2 * g;
        float bvals[2];
#pragma unroll
        for (int u = 0; u < 2; ++u) {
          int j = j0 + u;
          float v = 0.f;
          if (j < 18) {
            int t = j >> 1;
            int h2 = ho + t / 3 - 1;
            int w2 = wo + t % 3 - 1;
            bool ok = (h2 >= 0) & (h2 < HH) & (w2 >= 0) & (w2 < WW);
            float val = ((j & 1) == 0) ? ((float)h2 * inv128)
                                       : ((float)w2 * inv128 - 0.5f);
            v = ok ? val : 0.f;
          }
          bvals[u] = v;
        }
        v2f bv = {bvals[0], bvals[1]};
        int kk0 = 4 * s + 2 * g;
        v2f a0 = *(const v2f*)&s_w[((0 * 16 + px) << 6) + kk0];
        v2f a1 = *(const v2f*)&s_w[((1 * 16 + px) << 6) + kk0];
        WMMA_F32(acc0, a0, bv);
        WMMA_F32(acc1, a1, bv);
      }
    }
  }

  // Epilogue on C layout: channel m = mt*16 + r + 8*g, pixel = px
#pragma unroll
  for (int r = 0; r < 8; ++r) {
    {  // tile 0: channels 0..15, all idx channels
      int m = r + 8 * g;
      float om = acc0[r] + b_om[m];
      float base = (m < 9) ? ((float)ho * inv128 + (float)(m / 3 - 1))
                           : ((float)wo * inv128 - 0.5f + (float)((m - 9) % 3 - 1));
      out[IDX_OFF + (((b * 18 + m) * HH + ho) << 7) + wo] = base + om;
    }
    {  // tile 1: channels 16..31 (16,17 idx; 18..26 mask; 27..31 pad)
      int m = 16 + r + 8 * g;
      if (m < 18) {
        float om = acc1[r] + b_om[m];
        float base = (float)wo * inv128 - 0.5f + (float)((m - 9) % 3 - 1);
        out[IDX_OFF + (((b * 18 + m) * HH + ho) << 7) + wo] = base + om;
      } else if (m < CHOM) {
        float om = acc1[r] + b_om[m];
        out[MASK_OFF + (((b * 9 + (m - 18)) * HH + ho) << 7) + wo] =
            1.f / (1.f + __expf(-om));
      }
    }
  }
}

// ===========================================================================
// Kernel 2: deformable main GEMM via V_WMMA_F32_16X16X4_F32.
// Per wave: 16-pixel strip (N), 64 out channels (4 M-tiles), K tap-major.
// K-block = one tap: bilinear weights/corners hoisted out of the inner loop.
// ===========================================================================
__global__ __launch_bounds__(256) void dcn_main_kernel(
    const float* __restrict__ input, const float* __restrict__ weight,
    const float* __restrict__ bias, float* out) {
  __shared__ float s_w[64 * 64];             // weight chunk [m][kk], 16 KB
  __shared__ float s_tap[8 * 16 * 9 * 6];    // per-wave tap data, 27.6 KB

  const int tid  = threadIdx.x;
  const int wave = tid >> 5;
  const int lane = tid & 31;
  const int g    = lane >> 4;
  const int px   = lane & 15;

  const int strip = blockIdx.x * 8 + wave;   // 8192 strips total
  const int wtile = strip & 7;
  const int ho    = (strip >> 3) & 127;
  const int b     = strip >> 10;
  const int wo0   = wtile << 4;
  const float inv128 = 1.f / 128.f;

  // ---- per-wave tap precompute: 16 px * 9 taps, mask folded into weights
  for (int i = lane; i < 144; i += 32) {
    int p = i / 9;
    int t = i - p * 9;
    int wo = wo0 + p;
    int c0 = 2 * t, c1 = 2 * t + 1;
    float idx0 = out[IDX_OFF + (((b * 18 + c0) * HH + ho) << 7) + wo];
    float idx1 = out[IDX_OFF + (((b * 18 + c1) * HH + ho) << 7) + wo];
    float base0 = (c0 < 9) ? ((float)ho * inv128 + (float)(c0 / 3 - 1))
                           : ((float)wo * inv128 - 0.5f + (float)((c0 - 9) % 3 - 1));
    float base1 = (c1 < 9) ? ((float)ho * inv128 + (float)(c1 / 3 - 1))
                           : ((float)wo * inv128 - 0.5f + (float)((c1 - 9) % 3 - 1));
    float offh = idx0 - base0;
    float offw = idx1 - base1;
    float mk = out[MASK_OFF + (((b * 9 + t) * HH + ho) << 7) + wo];

    float ph = offh + (float)(t / 3) + (float)(ho - 1);
    float pw = offw + (float)(t % 3) + (float)(wo - 1);
    float h0f = floorf(ph), w0f = floorf(pw);
    float lh = ph - h0f, lw = pw - w0f;
    float hh = 1.f - lh, hw = 1.f - lw;
    int h0 = (int)h0f, w0 = (int)w0f;
    bool hin0 = (h0 >= 0) & (h0 < HH);
    bool hin1 = (h0 + 1 >= 0) & (h0 + 1 < HH);
    bool win0 = (w0 >= 0) & (w0 < WW);
    bool win1 = (w0 + 1 >= 0) & (w0 + 1 < WW);
    float* tp = &s_tap[((wave * 16 + p) * 9 + t) * 6];
    tp[0] = (hin0 & win0) ? hh * hw * mk : 0.f;
    tp[1] = (hin0 & win1) ? hh * lw * mk : 0.f;
    tp[2] = (hin1 & win0) ? lh * hw * mk : 0.f;
    tp[3] = (hin1 & win1) ? lh * lw * mk : 0.f;
    tp[4] = __int_as_float(h0);
    tp[5] = __int_as_float(w0);
  }

  v8f acc0 = {}; v8f acc1 = {}; v8f acc2 = {}; v8f acc3 = {};

  const float* tapBase = &s_tap[(wave * 16 + px) * 9 * 6];
  const float* inBase  = input + ((size_t)b * CIN << 14);

  for (int kb = 0; kb < 10; ++kb) {
    __syncthreads();
    // Stage permuted weight chunk: s_w row m, col kk holds weight for output m, channel c(kk), tap t(kk)
    for (int i2 = tid; i2 < 4096; i2 += 256) {
      int m = i2 >> 6, kk = i2 & 63;
      float v = 0.f;
      if (kb < 9)              v = weight[m * KDIM + kk * 9 + kb];
      else if (kk < 18)        v = weight[m * KDIM + (CIN + (kk & 1)) * 9 + (kk >> 1)];
      s_w[i2] = v;
    }
    __syncthreads();

    if (kb < 9) {
      // tap t = kb: load tap record once, hoist corners/weights
      const float* tp = tapBase + kb * 6;
      float m00 = tp[0], m01 = tp[1], m10 = tp[2], m11 = tp[3];
      int h0 = __float_as_int(tp[4]);
      int w0 = __float_as_int(tp[5]);
      int h0c = min(max(h0, 0), HH - 1);
      int h1c = min(max(h0 + 1, 0), HH - 1);
      int w0c = min(max(w0, 0), WW - 1);
      int w1c = min(max(w0 + 1, 0), WW - 1);
      int o00 = (h0c << 7) + w0c;
      int o01 = (h0c << 7) + w1c;
      int o10 = (h1c << 7) + w0c;
      int o11 = (h1c << 7) + w1c;
#pragma unroll 4
      for (int s = 0; s < 16; ++s) {
        int c0 = 4 * s + 2 * g;
        const float* xp0 = inBase + ((size_t)c0 << 14);
        const float* xp1 = xp0 + (1 << 14);
        float b0 = m00 * xp0[o00] + m01 * xp0[o01] + m10 * xp0[o10] + m11 * xp0[o11];
        float b1 = m00 * xp1[o00] + m01 * xp1[o01] + m10 * xp1[o10] + m11 * xp1[o11];
        v2f bv = {b0, b1};
        int kk0 = 4 * s + 2 * g;
        v2f a0 = *(const v2f*)&s_w[((0 * 16 + px) << 6) + kk0];
        v2f a1 = *(const v2f*)&s_w[((1 * 16 + px) << 6) + kk0];
        v2f a2 = *(const v2f*)&s_w[((2 * 16 + px) << 6) + kk0];
        v2f a3 = *(const v2f*)&s_w[((3 * 16 + px) << 6) + kk0];
        WMMA_F32(acc0, a0, bv);
        WMMA_F32(acc1, a1, bv);
        WMMA_F32(acc2, a2, bv);
        WMMA_F32(acc3, a3, bv);
      }
    } else {
      // analytic coord channels: j = 4s+2g+{0,1}, t=j>>1, c=64+(j&1); no vmem
      for (int s = 0; s < 16; ++s) {
        int j0 = 4 * s + 2 * g;
        float bvals[2];
#pragma unroll
        for (int u = 0; u < 2; ++u) {
          int j = j0 + u;
          float v = 0.f;
          if (j < 18) {
            int t = j >> 1;
            const float* tp = tapBase + t * 6;
            float m00 = tp[0], m01 = tp[1], m10 = tp[2], m11 = tp[3];
            int h0 = __float_as_int(tp[4]);
            int w0 = __float_as_int(tp[5]);
            int h0c = min(max(h0, 0), HH - 1);
            int h1c = min(max(h0 + 1, 0), HH - 1);
            int w0c = min(max(w0, 0), WW - 1);
            int w1c = min(max(w0 + 1, 0), WW - 1);
            if ((j & 1) == 0)
              v = (m00 + m01) * ((float)h0c * inv128) +
                  (m10 + m11) * ((float)h1c * inv128);
            else
              v = (m00 + m10) * ((float)w0c * inv128 - 0.5f) +
                  (m01 + m11) * ((float)w1c * inv128 - 0.5f);
          }
          bvals[u] = v;
        }
        v2f bv = {bvals[0], bvals[1]};
        int kk0 = 4 * s + 2 * g;
        v2f a0 = *(const v2f*)&s_w[((0 * 16 + px) << 6) + kk0];
        v2f a1 = *(const v2f*)&s_w[((1 * 16 + px) << 6) + kk0];
        v2f a2 = *(const v2f*)&s_w[((2 * 16 + px) << 6) + kk0];
        v2f a3 = *(const v2f*)&s_w[((3 * 16 + px) << 6) + kk0];
        WMMA_F32(acc0, a0, bv);
        WMMA_F32(acc1, a1, bv);
        WMMA_F32(acc2, a2, bv);
        WMMA_F32(acc3, a3, bv);
      }
    }
  }

  // Epilogue: C/D layout -> M = r + 8*g (+16*mt), N = px.  Add bias, store.
  const size_t pixoff = ((size_t)ho << 7) + wo0 + px;
#pragma unroll
  for (int r = 0; r < 8; ++r) {
    int m0 = r + 8 * g;
    out[OUT_OFF + (((size_t)b * COUT + (m0 +  0)) << 14) + pixoff] = acc0[r] + bias[m0 +  0];
    out[OUT_OFF + (((size_t)b * COUT + (m0 + 16)) << 14) + pixoff] = acc1[r] + bias[m0 + 16];
    out[OUT_OFF + (((size_t)b * COUT + (m0 + 32)) << 14) + pixoff] = acc2[r] + bias[m0 + 32];
    out[OUT_OFF + (((size_t)b * COUT + (m0 + 48)) << 14) + pixoff] = acc3[r] + bias[m0 + 48];
  }
}

// ---------------------------------------------------------------------------
extern "C" void kernel_launch(void* const* d_in, const int* in_sizes, int n_in,
                              void* d_out, int out_size, void* d_ws, size_t ws_size,
                              hipStream_t stream) {
  (void)in_sizes; (void)n_in; (void)out_size; (void)d_ws; (void)ws_size;
  const float* input  = (const float*)d_in[0];
  const float* weight = (const float*)d_in[1];
  const float* bias   = (const float*)d_in[2];
  const float* w_om   = (const float*)d_in[3];
  const float* b_om   = (const float*)d_in[4];
  float* out = (float*)d_out;

  // Kernel 1: om conv on the matrix pipe; writes idx + mask regions of d_out.
  dcn_om_wmma_kernel<<<1024, 256, 0, stream>>>(input, w_om, b_om, out);
  // Kernel 2: deformable main GEMM; reads idx/mask back, writes out region.
  dcn_main_kernel<<<1024, 256, 0, stream>>>(input, weight, bias, out);
}